// Model_936302870738
// MI455X (gfx1250) — compile-verified
//
#include <hip/hip_runtime.h>
#include <hip/hip_bf16.h>

// CDNA5 / gfx1250 wave32 WMMA types
typedef __attribute__((ext_vector_type(2))) float v2f;   // 16x4 f32 A / 4x16 f32 B fragment (2 VGPRs)
typedef __attribute__((ext_vector_type(8))) float v8f;   // 16x16 f32 C/D fragment (8 VGPRs)

#define B_   512
#define T_   1000
#define DI_  33
#define DR_  100
#define DO_  4

#define KTOT (DR_ + DI_)   // 133 combined K = [h | x]
#define KPAD 136           // round up to multiple of 4 (WMMA K step)
#define NK   (KPAD / 4)    // 34 wmma k-steps
#define ROW  138           // LDS row stride (floats): 69 8-byte units (odd) -> conflict-free b64 reads
#define NT   7             // n-tiles of 16 -> N = 112 >= DR_
#define BM   16            // batch rows per workgroup (one WMMA M-tile)
#define NTHREADS (NT * 32)

// retanh(x) = max(tanh(x),0) = tanh(max(x,0))  (tanh monotone, tanh(0)=0)
// For y >= 0:  tanh(y) = (1 - e^{-2y}) / (1 + e^{-2y}),  e^{-2y} = exp2(-2*log2e * y)
__device__ __forceinline__ float retanh_fast(float x) {
    float y = fmaxf(x, 0.0f);
    float e = __builtin_amdgcn_exp2f(-2.8853900817779268f * y);
    return (1.0f - e) * __builtin_amdgcn_rcpf(1.0f + e);
}

__global__ __launch_bounds__(NTHREADS) void rnn_scan_kernel(
    const float* __restrict__ input,   // [B,T,DI]
    const float* __restrict__ noise,   // [B,T,DR]
    const float* __restrict__ W_x,     // [DR,DI]
    const float* __restrict__ b_ah,    // [DR]
    const float* __restrict__ W_h,     // [DR,DR]
    const float* __restrict__ ah0,     // [DR]
    float* __restrict__ hstore)        // [B,T,DR]
{
    __shared__ float act[BM * ROW];    // [h(100) | x(33) | zero-pad] per batch row

    const int tid  = threadIdx.x;
    const int wave = tid >> 5;         // n-tile id 0..6
    const int lane = tid & 31;
    const int l16  = lane & 15;
    const int hi   = lane >> 4;        // 0/1
    const int koff = hi << 1;          // 0/2 : K sub-offset within a 4-wide fragment
    const int b0   = blockIdx.x * BM;

    const int  n      = wave * 16 + l16;   // output column (ah/h index), 0..111
    const bool nvalid = (n < DR_);

    // ---- preload combined-weight B fragments into registers (time-invariant) ----
    // B[k][n] = Wcomb[n][k],  Wcomb[n][k] = k<100 ? W_h[n][k] : (k<133 ? W_x[n][k-100] : 0)
    v2f bfrag[NK];
    #pragma unroll
    for (int kf = 0; kf < NK; ++kf) {
        const int k = kf * 4 + koff;
        float w0 = 0.0f, w1 = 0.0f;
        if (nvalid) {
            w0 = (k     < DR_) ? W_h[n * DR_ + k]
               : ((k     < KTOT) ? W_x[n * DI_ + (k     - DR_)] : 0.0f);
            w1 = (k + 1 < DR_) ? W_h[n * DR_ + k + 1]
               : ((k + 1 < KTOT) ? W_x[n * DI_ + (k + 1 - DR_)] : 0.0f);
        }
        bfrag[kf].x = w0;
        bfrag[kf].y = w1;
    }

    const float bias = nvalid ? b_ah[n] : 0.0f;

    // ---- persistent ah state in accumulator layout: lane -> N=n, vgpr r -> M = r + 8*hi ----
    float ah[8];
    const float a0 = nvalid ? ah0[n] : 0.0f;
    #pragma unroll
    for (int r = 0; r < 8; ++r) ah[r] = a0;

    // zero LDS (also establishes K padding region), then h0 = retanh(ah0), stage x(t=0)
    for (int i = tid; i < BM * ROW; i += NTHREADS) act[i] = 0.0f;
    __syncthreads();
    if (nvalid) {
        const float h0 = retanh_fast(a0);
        #pragma unroll
        for (int r = 0; r < 8; ++r) act[(r + 8 * hi) * ROW + n] = h0;
    }
    for (int i = tid; i < BM * DI_; i += NTHREADS) {
        const int m = i / DI_, j = i % DI_;
        act[m * ROW + DR_ + j] = input[(size_t)(b0 + m) * T_ * DI_ + j];
    }

    // per-lane base index into noise/hstore: ((b0 + 8*hi + r)*T + t)*DR + n
    const size_t rstride = (size_t)T_ * DR_;
    size_t base = (size_t)(b0 + 8 * hi) * rstride + (size_t)n;

    for (int t = 0; t < T_; ++t) {
        __syncthreads();   // h_t / x_t staged in act[]

        // issue noise loads for this step early (latency hidden behind WMMA chain)
        float nz[8];
        if (nvalid) {
            #pragma unroll
            for (int r = 0; r < 8; ++r) nz[r] = noise[base + (size_t)r * rstride];
        }
        // warm next step's noise lines
        if (nvalid && t + 1 < T_) {
            __builtin_prefetch(&noise[base + DR_], 0, 1);
            __builtin_prefetch(&noise[base + 4 * rstride + DR_], 0, 1);
        }

        // z[m][n] = sum_k act[m][k] * Wcomb[n][k]  -- 34 chained f32 WMMAs
        v8f acc = {};
        #pragma unroll
        for (int kf = 0; kf < NK; ++kf) {
            v2f a = *(const v2f*)&act[l16 * ROW + kf * 4 + koff];
            acc = __builtin_amdgcn_wmma_f32_16x16x4_f32(
                false, a, false, bfrag[kf], (short)0, acc, false, false);
        }

        // pointwise update + store h
        float h[8];
        #pragma unroll
        for (int r = 0; r < 8; ++r) {
            ah[r] = 0.9f * ah[r] + 0.1f * (acc[r] + bias);
            h[r]  = retanh_fast(ah[r]) + (nvalid ? nz[r] : 0.0f);
        }
        if (nvalid) {
            #pragma unroll
            for (int r = 0; r < 8; ++r) hstore[base + (size_t)r * rstride] = h[r];
        }

        __syncthreads();   // all A-fragment reads of old act done

        if (nvalid) {
            #pragma unroll
            for (int r = 0; r < 8; ++r) act[(r + 8 * hi) * ROW + n] = h[r];
        }
        if (t + 1 < T_) {
            for (int i = tid; i < BM * DI_; i += NTHREADS) {
                const int m = i / DI_, j = i % DI_;
                act[m * ROW + DR_ + j] =
                    input[(size_t)(b0 + m) * T_ * DI_ + (size_t)(t + 1) * DI_ + j];
            }
        }
        base += DR_;
    }
}

// out[b,t,:] = hstore[b,t,:] @ W_y.T + b_y   (DO=4, memory-bound pass)
__global__ __launch_bounds__(256) void out_proj_kernel(
    const float* __restrict__ hstore,  // [B,T,DR]
    const float* __restrict__ W_y,     // [DO,DR]
    const float* __restrict__ b_y,     // [DO]
    float* __restrict__ out)           // [B,T,DO]
{
    __shared__ float wy[DO_ * DR_];
    __shared__ float by[DO_];
    const int tid = threadIdx.x;
    for (int i = tid; i < DO_ * DR_; i += 256) wy[i] = W_y[i];
    if (tid < DO_) by[tid] = b_y[tid];
    __syncthreads();

    const size_t bt = (size_t)blockIdx.x * 256 + tid;
    if (bt >= (size_t)B_ * T_) return;
    const float* __restrict__ hrow = hstore + bt * DR_;

    float a0 = by[0], a1 = by[1], a2 = by[2], a3 = by[3];
    #pragma unroll 4
    for (int r = 0; r < DR_; ++r) {
        const float hv = hrow[r];
        a0 += hv * wy[0 * DR_ + r];
        a1 += hv * wy[1 * DR_ + r];
        a2 += hv * wy[2 * DR_ + r];
        a3 += hv * wy[3 * DR_ + r];
    }
    float* o = out + bt * DO_;
    o[0] = a0; o[1] = a1; o[2] = a2; o[3] = a3;
}

extern "C" void kernel_launch(void* const* d_in, const int* in_sizes, int n_in,
                              void* d_out, int out_size, void* d_ws, size_t ws_size,
                              hipStream_t stream) {
    (void)in_sizes; (void)n_in; (void)out_size; (void)d_ws; (void)ws_size;
    // setup_inputs() order:
    const float* input  = (const float*)d_in[0];  // [B,T,DI]
    const float* noise  = (const float*)d_in[1];  // [B,T,DR]
    const float* W_x_ah = (const float*)d_in[2];  // [DR,DI]
    const float* b_ah   = (const float*)d_in[3];  // [DR]
    const float* W_h_ah = (const float*)d_in[4];  // [DR,DR]
    const float* W_h_y  = (const float*)d_in[5];  // [DO,DR]
    const float* b_y    = (const float*)d_in[6];  // [DO]
    const float* ah0    = (const float*)d_in[7];  // [DR]

    float* out    = (float*)d_out;                         // [B,T,DO]
    float* hstore = (float*)d_out + (size_t)B_ * T_ * DO_; // [B,T,DR]

    rnn_scan_kernel<<<B_ / BM, NTHREADS, 0, stream>>>(
        input, noise, W_x_ah, b_ah, W_h_ah, ah0, hstore);

    const int nbt = (B_ * T_ + 255) / 256;
    out_proj_kernel<<<nbt, 256, 0, stream>>>(hstore, W_h_y, b_y, out);
}